// SpatialTransformer_27427661152748
// MI455X (gfx1250) — compile-verified
//
#include <hip/hip_runtime.h>

// ---- problem constants (from reference) ----
#define OHW   256
#define NPIX  65536        // 256*256 pixels per batch
#define BATCH 16
#define CH    64           // channels

typedef __attribute__((ext_vector_type(2))) float v2f;
typedef __attribute__((ext_vector_type(4))) float f4;
typedef __attribute__((ext_vector_type(8))) float v8f;

// -----------------------------------------------------------------------------
// Pass 1: affine grid generation via V_WMMA_F32_16X16X4_F32.
//   D[b, n] = sum_k A[b,k] * B[k,n],  K = {gx, gy, 1, 0-pad}
// One wave handles one 16-pixel tile for ALL 16 batches (x and y = 2 WMMAs).
// A-matrix layout (16x4 f32, 2 VGPRs): lanes 0-15 -> K=0,1 ; lanes 16-31 -> K=2,3
// B-matrix layout (4x16 f32, 2 VGPRs): lanes 0-15 -> rows K=0,1 ; lanes 16-31 -> rows K=2,3
// D layout (16x16 f32, 8 VGPRs): lane L<16, VGPR j -> (M=j, N=L); L>=16 -> (M=j+8, N=L-16)
// -----------------------------------------------------------------------------
__global__ void __launch_bounds__(256) st_grid_wmma(const float* __restrict__ theta,
                                                    float* __restrict__ xs_ws,
                                                    float* __restrict__ ys_ws) {
    const int lane   = threadIdx.x & 31;
    const int wave   = threadIdx.x >> 5;
    const int tile   = blockIdx.x * 8 + wave;   // 0..4095 (4096 tiles of 16 pixels)
    const int n0     = tile << 4;               // first pixel of tile (tile within one row)
    const int oy     = n0 >> 8;
    const int ox0    = n0 & 255;
    const int lane15 = lane & 15;
    const bool hi    = lane >= 16;

    // B matrix columns = pixels n0..n0+15 ; rows = [gx, gy, 1, 0]
    const float gx = fmaf((float)(ox0 + lane15), 2.0f / 255.0f, -1.0f);
    const float gy = fmaf((float)oy,             2.0f / 255.0f, -1.0f);
    v2f bmat;
    bmat.x = hi ? 1.0f : gx;
    bmat.y = hi ? 0.0f : gy;

    // A matrices: rows = batches, cols = theta coeffs (K=3 padded with 0)
    const float* t = theta + lane15 * 6;
    const float tx0 = t[0], tx1 = t[1], tx2 = t[2];
    const float ty0 = t[3], ty1 = t[4], ty2 = t[5];
    v2f ax, ay;
    ax.x = hi ? tx2 : tx0;  ax.y = hi ? 0.0f : tx1;
    ay.x = hi ? ty2 : ty0;  ay.y = hi ? 0.0f : ty1;

    v8f zero = {};
    v8f dx = __builtin_amdgcn_wmma_f32_16x16x4_f32(false, ax, false, bmat,
                                                   (short)0, zero, false, false);
    v8f dy = __builtin_amdgcn_wmma_f32_16x16x4_f32(false, ay, false, bmat,
                                                   (short)0, zero, false, false);

    // x = 0.5*(tg+1)*256 = 128*tg + 128  (same for y since H == W == 256)
    const int n    = n0 + lane15;
    const int brow = hi ? 8 : 0;
#pragma unroll
    for (int j = 0; j < 8; ++j) {
        const int b = brow + j;
        xs_ws[b * NPIX + n] = fmaf(dx[j], 128.0f, 128.0f);
        ys_ws[b * NPIX + n] = fmaf(dy[j], 128.0f, 128.0f);
    }
}

// -----------------------------------------------------------------------------
// Pass 2: bilinear gather. One thread = (pixel, channel-quad).
// 16 consecutive lanes cover one pixel's 64 channels -> each corner gather is a
// coalesced 256B run per half-wave (global_load_b128). Output written with
// non-temporal b128 stores to preserve L2 residency of the image.
// -----------------------------------------------------------------------------
template <bool USE_WS>
__global__ void __launch_bounds__(256) st_sample(const float* __restrict__ image,
                                                 const float* __restrict__ theta,
                                                 const float* __restrict__ xs_ws,
                                                 const float* __restrict__ ys_ws,
                                                 float* __restrict__ out) {
    const unsigned tid = blockIdx.x * 256u + threadIdx.x;
    const int cq  = (int)(tid & 15);          // which float4 of the 64 channels
    const int pix = (int)(tid >> 4);          // global pixel 0..B*NPIX-1
    const int b   = pix >> 16;                // NPIX == 1<<16
    const int n   = pix & (NPIX - 1);

    float x, y;
    if (USE_WS) {
        x = xs_ws[b * NPIX + n];
        y = ys_ws[b * NPIX + n];
    } else {
        const int oy = n >> 8;
        const int ox = n & 255;
        const float gx = fmaf((float)ox, 2.0f / 255.0f, -1.0f);
        const float gy = fmaf((float)oy, 2.0f / 255.0f, -1.0f);
        const float* t = theta + b * 6;
        x = fmaf(fmaf(t[0], gx, fmaf(t[1], gy, t[2])), 128.0f, 128.0f);
        y = fmaf(fmaf(t[3], gx, fmaf(t[4], gy, t[5])), 128.0f, 128.0f);
    }

    const float xf = floorf(x);
    const float yf = floorf(y);
    const int x0 = min(max((int)xf,     0), OHW - 1);
    const int x1 = min(max((int)xf + 1, 0), OHW - 1);
    const int y0 = min(max((int)yf,     0), OHW - 1);
    const int y1 = min(max((int)yf + 1, 0), OHW - 1);

    // weights use CLIPPED corner coords, exactly as the reference does
    const float x0f = (float)x0, x1f = (float)x1;
    const float y0f = (float)y0, y1f = (float)y1;
    const float wa = (x1f - x) * (y1f - y);
    const float wb = (x1f - x) * (y - y0f);
    const float wc = (x - x0f) * (y1f - y);
    const float wd = (x - x0f) * (y - y0f);

    // image[b, y, x, c] as float4 chunks: index = ((b*NPIX + y*256 + x) * 16) + cq
    const f4* img = (const f4*)image + ((size_t)b * NPIX) * 16 + cq;
    const f4 Ia = img[(y0 * OHW + x0) * 16];
    const f4 Ib = img[(y1 * OHW + x0) * 16];
    const f4 Ic = img[(y0 * OHW + x1) * 16];
    const f4 Id = img[(y1 * OHW + x1) * 16];

    const f4 r = wa * Ia + wb * Ib + wc * Ic + wd * Id;

    f4* op = (f4*)out + (size_t)pix * 16 + cq;
    __builtin_nontemporal_store(r, op);   // keep L2 for the gathered image
}

extern "C" void kernel_launch(void* const* d_in, const int* in_sizes, int n_in,
                              void* d_out, int out_size, void* d_ws, size_t ws_size,
                              hipStream_t stream) {
    const float* image = (const float*)d_in[0];   // [16,256,256,64] f32
    const float* theta = (const float*)d_in[1];   // [16,6] f32
    float* out = (float*)d_out;                   // [16,256,256,64] f32

    const size_t coord_elems = (size_t)BATCH * NPIX;          // per array
    const size_t ws_need     = 2 * coord_elems * sizeof(float); // 8 MB

    const unsigned nThreads = (unsigned)BATCH * NPIX * 16u;   // 16,777,216
    dim3 sblock(256), sgrid(nThreads / 256);                  // 65536 blocks

    if (ws_size >= ws_need) {
        float* xs_ws = (float*)d_ws;
        float* ys_ws = xs_ws + coord_elems;
        // 4096 tiles, 8 waves per 256-thread block -> 512 blocks
        st_grid_wmma<<<512, 256, 0, stream>>>(theta, xs_ws, ys_ws);
        st_sample<true><<<sgrid, sblock, 0, stream>>>(image, theta, xs_ws, ys_ws, out);
    } else {
        st_sample<false><<<sgrid, sblock, 0, stream>>>(image, theta, nullptr, nullptr, out);
    }
}